// LaterallyConnectedLayer_13864154432178
// MI455X (gfx1250) — compile-verified
//
#include <hip/hip_runtime.h>

// LaterallyConnectedLayer for MI455X (gfx1250), fp32 WMMA path.
//
// Simplified math (see analysis): both lateral passes are implicit-GEMM 5x5
// convs run on V_WMMA_F32_16X16X4_F32; pass-1 folds the 4x channel tiling
// into the weights; pass-2 is a winner-gathered 64->64 conv per sample.

typedef float v2f __attribute__((ext_vector_type(2)));
typedef float v8f __attribute__((ext_vector_type(8)));

#define PADW 36          // 2+32+2 padded width/height
#define K1   1600        // 64 in-channels * 25 taps

// workspace layout (float offsets)
#define OFF_APAD 0                       // 16*64*36*36      = 1327104
#define OFF_WN   1327104                 // 256*256*25       = 1638400
#define OFF_W1S  (OFF_WN + 1638400)      // 256*64*25        = 409600
#define OFF_L1   (OFF_W1S + 409600)      // 16*256*1024      = 4194304
#define OFF_SL1  (OFF_L1 + 4194304)      // 16*256           = 4096
#define OFF_WIN  (OFF_SL1 + 4096)        // 16*64 ints

// ---- Wn[o,i,r] = minmax over 25 taps of K[i,o,r] -----------------------
__global__ void lcl_normalize_K(const float* __restrict__ K, float* __restrict__ Wn) {
    int m = blockIdx.x * blockDim.x + threadIdx.x;   // 65536 maps, m = o*256+i
    int o = m >> 8, i = m & 255;
    const float* src = K + ((size_t)i * 256 + o) * 25;
    float mn = src[0], mx = src[0];
#pragma unroll
    for (int r = 1; r < 25; ++r) { float v = src[r]; mn = fminf(mn, v); mx = fmaxf(mx, v); }
    float d = mx - mn;
    float* dst = Wn + (size_t)m * 25;
    if (d > 0.f) {
        float inv = 1.0f / d;
#pragma unroll
        for (int r = 0; r < 25; ++r) dst[r] = (src[r] - mn) * inv;
    } else {
#pragma unroll
        for (int r = 0; r < 25; ++r) dst[r] = 0.f;
    }
}

// ---- W1s[c,f,r] = sum_n Wn[c, n*64+f, r]  (fold the 4x channel tiling) --
__global__ void lcl_build_w1s(const float* __restrict__ Wn, float* __restrict__ W1s) {
    int t = blockIdx.x * blockDim.x + threadIdx.x;   // 409600
    int r = t % 25; int cf = t / 25;
    int f = cf & 63; int c = cf >> 6;
    float s = 0.f;
#pragma unroll
    for (int n = 0; n < 4; ++n)
        s += Wn[((size_t)c * 256 + n * 64 + f) * 25 + r];
    W1s[t] = s;                                      // layout (c*64+f)*25+r
}

// ---- zero-padded activations: Apad[b,f, 36,36], A at [2:34,2:34] -------
__global__ void lcl_pad_A(const float* __restrict__ A, float* __restrict__ Apad) {
    int t = blockIdx.x * blockDim.x + threadIdx.x;   // 16*64*36*36 = 1327104
    int x = t % PADW; int rest = t / PADW;
    int y = rest % PADW; int bc = rest / PADW;
    float v = 0.f;
    if (x >= 2 && x < 34 && y >= 2 && y < 34)
        v = A[(size_t)bc * 1024 + (y - 2) * 32 + (x - 2)];
    Apad[t] = v;
}

// ---- pass 1: L1raw[b,c,p] = W1s[c] (x) Apad[b]  (implicit-GEMM WMMA) ----
// M=256 out-channels, N=16x1024 pixels, K=1600. One wave -> 16x64 tile.
__global__ void __launch_bounds__(256) lcl_pass1(
    const float* __restrict__ W1s, const float* __restrict__ Apad,
    float* __restrict__ L1raw)
{
    const int lane  = threadIdx.x & 31;
    const int wave  = blockIdx.x * 8 + (threadIdx.x >> 5);  // 4096 waves exactly
    const int mtile = wave & 15;
    const int ntile = wave >> 4;          // 256 pixel tiles (64 px each)
    const int b     = ntile >> 4;
    const int y0    = (ntile & 15) << 1;  // two full rows per wave
    const int c0    = mtile << 4;

    const int col   = lane & 15;          // A-row m / B-col n
    const int half  = lane >> 4;
    const int kHalf = half << 1;

    v8f acc[4] = {};
    const float* wrow = W1s + (size_t)(c0 + col) * K1 + kHalf;
    const float* apb  = Apad + (size_t)b * (64 * PADW * PADW);

#pragma unroll 2
    for (int kb = 0; kb < K1; kb += 4) {
        v2f a = *(const v2f*)(wrow + kb);             // A frag: W1s[m, kb+kHalf..+1]
        v2f bf[4];
#pragma unroll
        for (int t = 0; t < 2; ++t) {                 // B frag: implicit im2col
            int k  = kb + kHalf + t;
            int f  = k / 25;
            int r  = k - f * 25;
            int ky = r / 5;
            int kx = r - ky * 5;
            const float* src = apb + (size_t)(f * PADW + y0 + ky) * PADW + kx + col;
#pragma unroll
            for (int j = 0; j < 4; ++j) {
                float v = src[(j >> 1) * PADW + (j & 1) * 16];
                if (t == 0) bf[j].x = v; else bf[j].y = v;
            }
        }
#pragma unroll
        for (int j = 0; j < 4; ++j)
            acc[j] = __builtin_amdgcn_wmma_f32_16x16x4_f32(
                false, a, false, bf[j], (short)0, acc[j], false, false);
    }

    float* out = L1raw + ((size_t)(b * 256 + c0) << 10);
#pragma unroll
    for (int j = 0; j < 4; ++j) {
        int p = (y0 + (j >> 1)) * 32 + (j & 1) * 16 + col;
#pragma unroll
        for (int r = 0; r < 8; ++r)
            out[((size_t)(half * 8 + r) << 10) + p] = acc[j][r];
    }
}

// ---- fused per-map min/max + normalized-sum (wave32 shuffles) -----------
// sL1[b,c] = sum(minmax(conv*S)) ; /64 dropped (minmax scale-invariant)
__global__ void __launch_bounds__(256) lcl_reduce(
    const float* __restrict__ L1raw, const float* __restrict__ S,
    float* __restrict__ sL1)
{
    const int lane = threadIdx.x & 31;
    const int map  = blockIdx.x * 8 + (threadIdx.x >> 5);   // 4096 maps
    const float s  = S[map & 255];
    const float* base = L1raw + ((size_t)map << 10);

    float mn = 3.402823466e38f, mx = -3.402823466e38f;
#pragma unroll 4
    for (int i = 0; i < 32; ++i) {
        float v = base[lane + (i << 5)] * s;
        mn = fminf(mn, v); mx = fmaxf(mx, v);
    }
#pragma unroll
    for (int off = 16; off > 0; off >>= 1) {
        mn = fminf(mn, __shfl_xor(mn, off, 32));
        mx = fmaxf(mx, __shfl_xor(mx, off, 32));
    }
    float denom = mx - mn;
    float local = 0.f;
#pragma unroll 4
    for (int i = 0; i < 32; ++i)
        local += base[lane + (i << 5)] * s - mn;
#pragma unroll
    for (int off = 16; off > 0; off >>= 1)
        local += __shfl_xor(local, off, 32);
    if (lane == 0)
        sL1[map] = (denom > 0.f) ? (local / denom) : 0.f;
}

// ---- winner per (b, feature-map): argmax over 4 multiplex copies --------
// (0.9*sumA term identical across copies -> drops out of the comparison)
__global__ void lcl_argmax(const float* __restrict__ sL1, int* __restrict__ win) {
    int t = blockIdx.x * blockDim.x + threadIdx.x;   // 1024
    int b = t >> 6, f = t & 63;
    float best = sL1[b * 256 + f]; int bn = 0;
#pragma unroll
    for (int n = 1; n < 4; ++n) {
        float v = sL1[b * 256 + n * 64 + f];
        if (v > best) { best = v; bn = n; }          // strict > keeps first (argmax)
    }
    win[t] = bn * 64 + f;
}

// ---- pass 2: out[b,f] = (1/64) sum_f' Wn[win[f],win[f']] (x) A[b,f'] ----
__global__ void __launch_bounds__(256) lcl_pass2(
    const float* __restrict__ Wn, const float* __restrict__ Apad,
    const int* __restrict__ win, float* __restrict__ out)
{
    const int lane  = threadIdx.x & 31;
    const int wave  = blockIdx.x * 8 + (threadIdx.x >> 5);  // 1024 waves exactly
    const int b     = wave >> 6;
    const int rem   = wave & 63;
    const int mtile = rem & 3;
    const int ntile = rem >> 2;
    const int y0    = ntile << 1;
    const int c0    = mtile << 4;

    const int col   = lane & 15;
    const int half  = lane >> 4;
    const int kHalf = half << 1;

    const int* winb = win + b * 64;
    const int  wo   = winb[c0 + col];     // gathered out-channel for lane's M row

    v8f acc[4] = {};
    const float* apb = Apad + (size_t)b * (64 * PADW * PADW);

#pragma unroll 2
    for (int kb = 0; kb < K1; kb += 4) {
        v2f a, bf[4];
#pragma unroll
        for (int t = 0; t < 2; ++t) {
            int k  = kb + kHalf + t;
            int f  = k / 25;
            int r  = k - f * 25;
            int wi = winb[f];
            float wv = Wn[((size_t)wo * 256 + wi) * 25 + r];   // gathered weights
            if (t == 0) a.x = wv; else a.y = wv;
            int ky = r / 5;
            int kx = r - ky * 5;
            const float* src = apb + (size_t)(f * PADW + y0 + ky) * PADW + kx + col;
#pragma unroll
            for (int j = 0; j < 4; ++j) {
                float v = src[(j >> 1) * PADW + (j & 1) * 16];
                if (t == 0) bf[j].x = v; else bf[j].y = v;
            }
        }
#pragma unroll
        for (int j = 0; j < 4; ++j)
            acc[j] = __builtin_amdgcn_wmma_f32_16x16x4_f32(
                false, a, false, bf[j], (short)0, acc[j], false, false);
    }

    float* ob = out + ((size_t)(b * 64 + c0) << 10);
#pragma unroll
    for (int j = 0; j < 4; ++j) {
        int p = (y0 + (j >> 1)) * 32 + (j & 1) * 16 + col;
#pragma unroll
        for (int r = 0; r < 8; ++r)
            ob[((size_t)(half * 8 + r) << 10) + p] = acc[j][r] * 0.015625f;  // /64
    }
}

extern "C" void kernel_launch(void* const* d_in, const int* in_sizes, int n_in,
                              void* d_out, int out_size, void* d_ws, size_t ws_size,
                              hipStream_t stream)
{
    const float* A = (const float*)d_in[0];   // (16,64,32,32)
    const float* K = (const float*)d_in[1];   // (256,256,5,5)
    const float* S = (const float*)d_in[2];   // (256,)
    float* ws   = (float*)d_ws;               // ~30.3 MB used
    float* Apad = ws + OFF_APAD;
    float* Wn   = ws + OFF_WN;
    float* W1s  = ws + OFF_W1S;
    float* L1   = ws + OFF_L1;
    float* sL1  = ws + OFF_SL1;
    int*   win  = (int*)(ws + OFF_WIN);
    float* outp = (float*)d_out;              // (16,64,32,32)

    lcl_normalize_K<<<dim3(256),  dim3(256), 0, stream>>>(K, Wn);
    lcl_build_w1s  <<<dim3(1600), dim3(256), 0, stream>>>(Wn, W1s);
    lcl_pad_A      <<<dim3(5184), dim3(256), 0, stream>>>(A, Apad);
    lcl_pass1      <<<dim3(512),  dim3(256), 0, stream>>>(W1s, Apad, L1);
    lcl_reduce     <<<dim3(512),  dim3(256), 0, stream>>>(L1, S, sL1);
    lcl_argmax     <<<dim3(4),    dim3(256), 0, stream>>>(sL1, win);
    lcl_pass2      <<<dim3(128),  dim3(256), 0, stream>>>(Wn, Apad, win, outp);
}